// VAE_16475494548020
// MI455X (gfx1250) — compile-verified
//
#include <hip/hip_runtime.h>

#define VOCAB 32128
#define NHID  16
#define OUT   16
#define BATCH 4096
#define SEQ   200

typedef float v2f __attribute__((ext_vector_type(2)));
typedef float v8f __attribute__((ext_vector_type(8)));

// ---------------------------------------------------------------------------
// Kernel 1: embedding-bag + ReLU + mean/logvar heads + reparameterization.
// Block = 256 threads = 16 rows x 16 hidden units.
// ---------------------------------------------------------------------------
__global__ __launch_bounds__(256) void vae_front_kernel(
    const int*   __restrict__ x,        // [B, S]
    const float* __restrict__ eps,      // [B, 16]
    const float* __restrict__ enc_w,    // [V, 16]
    const float* __restrict__ enc_b,    // [16]
    const float* __restrict__ mean_w,   // [16, 16]
    const float* __restrict__ mean_b,   // [16]
    const float* __restrict__ logvar_w, // [16, 16]
    const float* __restrict__ logvar_b, // [16]
    float* __restrict__ mean_out,       // [B, 16]
    float* __restrict__ logvar_out,     // [B, 16]
    float* __restrict__ z_out)          // [B, 16] (workspace)
{
    const int tid = threadIdx.x;
    const int r   = tid >> 4;                  // row within block (0..15)
    const int j   = tid & 15;                  // hidden index (0..15)
    const int row = blockIdx.x * 16 + r;

    // Embedding-bag: h[row][j] = relu(enc_b[j] + sum_s enc_w[x[row][s]][j])
    const int* xr = x + row * SEQ;
    float acc = enc_b[j];
    #pragma unroll 5
    for (int s = 0; s < SEQ; s += 4) {         // SEQ=200 divisible by 4
        int4 t = *reinterpret_cast<const int4*>(xr + s);
        acc += enc_w[t.x * NHID + j];          // 16 lanes hit one 64B row
        acc += enc_w[t.y * NHID + j];
        acc += enc_w[t.z * NHID + j];
        acc += enc_w[t.w * NHID + j];
    }
    const float h = fmaxf(acc, 0.0f);

    __shared__ float hs[16][NHID + 1];         // +1 pad: avoid bank conflicts
    hs[r][j] = h;
    __syncthreads();

    float m  = mean_b[j];
    float lv = logvar_b[j];
    #pragma unroll
    for (int k = 0; k < NHID; ++k) {
        const float hk = hs[r][k];
        m  = fmaf(hk, mean_w[k * OUT + j],   m);
        lv = fmaf(hk, logvar_w[k * OUT + j], lv);
    }

    const float z = m + eps[row * OUT + j] * (0.5f * __expf(lv));

    mean_out[row * OUT + j]   = m;
    logvar_out[row * OUT + j] = lv;
    z_out[row * OUT + j]      = z;
}

// ---------------------------------------------------------------------------
// Kernel 2: dec = sigmoid(relu(z @ dec_w + dec_b))  via f32 WMMA.
// One wave computes one 16x16 tile of dec; K=16 split into 4 WMMA k-steps.
// Block = 128 threads = 4 waves covering 4 adjacent V-tiles (same B-tile,
// so the 1KB z tile is shared through L0).
// Roofline: dec output is 526 MB -> ~23 us at 23.3 TB/s; WMMA supplies the
// 186 TFLOP/s f32 needed at that rate, leaving ~5 VALU ops/element for the
// bias+relu+sigmoid epilogue (hardware v_rcp_f32, no IEEE-div expansion).
// ---------------------------------------------------------------------------
__global__ __launch_bounds__(128) void vae_dec_kernel(
    const float* __restrict__ z,      // [B, 16]
    const float* __restrict__ dec_w,  // [16, V]
    const float* __restrict__ dec_b,  // [V]
    float* __restrict__ dec_out)      // [B, V]
{
    const int lane  = threadIdx.x & 31;
    const int wave  = threadIdx.x >> 5;
    const int vtile = blockIdx.x * 4 + wave;
    const int btile = blockIdx.y;
    const int v0 = vtile * 16;
    const int b0 = btile * 16;

    const int n    = lane & 15;   // N index (B/C/D) and M index (A)
    const int half = lane >> 4;   // 0 -> K pair {k0,k0+1}; 1 -> {k0+2,k0+3}

    const float* zrow = z + (b0 + n) * OUT;   // A: M = lane%16, row-major z
    const float* wcol = dec_w + v0 + n;       // B: N = lane%16, row-major dec_w

    v8f acc = {};
    #pragma unroll
    for (int k0 = 0; k0 < OUT; k0 += 4) {
        const int ka = k0 + 2 * half;
        v2f a, b;
        a.x = zrow[ka];                        // A VGPR0: K=ka
        a.y = zrow[ka + 1];                    // A VGPR1: K=ka+1
        b.x = wcol[(size_t)ka * VOCAB];        // B VGPR0: K=ka (16 lanes = 64B)
        b.y = wcol[(size_t)(ka + 1) * VOCAB];  // B VGPR1: K=ka+1
        // D = A(16x4) * B(4x16) + C ; 8 args: neg_a,A,neg_b,B,c_mod,C,reuse_a,reuse_b
        acc = __builtin_amdgcn_wmma_f32_16x16x4_f32(
            false, a, false, b, (short)0, acc, false, false);
    }

    // Epilogue: bias + relu + sigmoid (single v_rcp_f32, not IEEE divide),
    // then streaming (non-temporal) stores — dec (526 MB) > L2 (192 MB),
    // keep it from evicting the L2-resident dec_w / z working set.
    const float bias = dec_b[v0 + n];
    float* outcol = dec_out + (size_t)b0 * VOCAB + v0 + n;
    #pragma unroll
    for (int i = 0; i < 8; ++i) {
        // C/D layout: VGPR i -> M = i + 8*half, N = lane%16
        float t = fmaxf(acc[i] + bias, 0.0f);
        float s = __builtin_amdgcn_rcpf(1.0f + __expf(-t));
        __builtin_nontemporal_store(s, outcol + (size_t)(i + 8 * half) * VOCAB);
    }
}

// ---------------------------------------------------------------------------
extern "C" void kernel_launch(void* const* d_in, const int* in_sizes, int n_in,
                              void* d_out, int out_size, void* d_ws, size_t ws_size,
                              hipStream_t stream) {
    const int*   x        = (const int*)  d_in[0];
    const float* eps      = (const float*)d_in[1];
    const float* enc_w    = (const float*)d_in[2];
    const float* enc_b    = (const float*)d_in[3];
    const float* mean_w   = (const float*)d_in[4];
    const float* mean_b   = (const float*)d_in[5];
    const float* logvar_w = (const float*)d_in[6];
    const float* logvar_b = (const float*)d_in[7];
    const float* dec_w    = (const float*)d_in[8];
    const float* dec_b    = (const float*)d_in[9];

    float* out        = (float*)d_out;
    float* dec_out    = out;                                  // [B, V]
    float* mean_out   = out + (size_t)BATCH * VOCAB;          // [B, 16]
    float* logvar_out = mean_out + (size_t)BATCH * OUT;       // [B, 16]
    float* z_ws       = (float*)d_ws;                         // [B, 16]

    vae_front_kernel<<<dim3(BATCH / 16), dim3(256), 0, stream>>>(
        x, eps, enc_w, enc_b, mean_w, mean_b, logvar_w, logvar_b,
        mean_out, logvar_out, z_ws);

    // V/16 = 2008 tiles, 4 per block -> 502 blocks in x; B/16 = 256 in y.
    vae_dec_kernel<<<dim3(VOCAB / 16 / 4, BATCH / 16), dim3(128), 0, stream>>>(
        z_ws, dec_w, dec_b, dec_out);
}